// TripletLoss_Mixup_76991583748406
// MI455X (gfx1250) — compile-verified
//
#include <hip/hip_runtime.h>

typedef __attribute__((ext_vector_type(2))) float v2f;
typedef __attribute__((ext_vector_type(8))) float v8f;

#define NROWS 4096
#define NDIM  2048
#define MARGIN 0.3f

#define BM 128           // block rows
#define BN 64            // block cols
#define KC 32            // K chunk
#define KPAD 36          // LDS row stride (mult of 4 for b128; 9r%16 distinct -> conflict-free)
#define NCHUNK (NDIM / KC)

// ---------------------------------------------------------------------------
// gfx1250 async global->LDS copy (ASYNCcnt-tracked DMA, ISA 15.18.3 op 98).
// asm volatile + memory clobber pins the issue point: copies go out at the
// top of the loop body and are only waited on after the whole compute phase.
// ---------------------------------------------------------------------------
__device__ __forceinline__ void async_copy_b128(unsigned lds_off,
                                                const float* gsrc) {
  asm volatile("global_load_async_to_lds_b128 %0, %1, off"
               :: "v"(lds_off), "v"((unsigned long long)(uintptr_t)gsrc)
               : "memory");
}
__device__ __forceinline__ void wait_async0() {
  asm volatile("s_wait_asynccnt 0x0" ::: "memory");
}

// ---------------------------------------------------------------------------
// Init per-row mining stats (positive floats: uint bits are order-preserving).
// ---------------------------------------------------------------------------
__global__ __launch_bounds__(256) void init_stats(unsigned* __restrict__ hp,
                                                  unsigned* __restrict__ hn,
                                                  unsigned* __restrict__ mo,
                                                  unsigned* __restrict__ mn,
                                                  unsigned* __restrict__ has) {
  int i = blockIdx.x * 256 + threadIdx.x;
  if (i < NROWS) {
    hp[i]  = 0u;
    hn[i]  = 0x7F800000u;
    mo[i]  = 0u;
    mn[i]  = 0x7F800000u;
    has[i] = 0u;
  }
}

// ---------------------------------------------------------------------------
// sq[i] = sum_k X[i,k]^2
// ---------------------------------------------------------------------------
__global__ __launch_bounds__(256) void sqnorm(const float* __restrict__ X,
                                              float* __restrict__ sq) {
  __shared__ float red[256];
  const int row = blockIdx.x;
  const float* xr = X + (size_t)row * NDIM;
  float s = 0.f;
  for (int k = threadIdx.x * 4; k < NDIM; k += 256 * 4) {
    float4 v = *(const float4*)(xr + k);
    s += v.x * v.x + v.y * v.y + v.z * v.z + v.w * v.w;
  }
  red[threadIdx.x] = s;
  __syncthreads();
  for (int off = 128; off > 0; off >>= 1) {
    if (threadIdx.x < off) red[threadIdx.x] += red[threadIdx.x + off];
    __syncthreads();
  }
  if (threadIdx.x == 0) sq[row] = red[0];
}

// ---------------------------------------------------------------------------
// Async-copy double-buffered Gram GEMM (X X^T) with V_WMMA_F32_16X16X4_F32,
// fused distance + masked hardest-pos/neg mining.
// Block: 8 waves -> 128x64 output; wave (w>>1, w&1) owns a 32x32 sub-tile
// (2 A frags x 2 B frags -> 4 WMMAs per K=4 step).
// ---------------------------------------------------------------------------
__global__ __launch_bounds__(256) void gram_stats(
    const float* __restrict__ X, const float* __restrict__ sq,
    const int* __restrict__ ta, const int* __restrict__ tb,
    unsigned* __restrict__ hp, unsigned* __restrict__ hn,
    unsigned* __restrict__ mo, unsigned* __restrict__ mn,
    unsigned* __restrict__ has) {
  __shared__ float lds[2 * BM * KPAD + 2 * BN * KPAD];
  float* ldsA = lds;                        // [2][BM][KPAD]
  float* ldsB = lds + 2 * BM * KPAD;        // [2][BN][KPAD]

  const int tid  = threadIdx.x;
  const int lane = tid & 31;
  const int wave = tid >> 5;
  const int wm   = wave >> 1;               // 0..3 -> 32-row band
  const int wn   = wave & 1;                // 0..1 -> 32-col band
  const int l16  = lane & 15;
  const int koff = (lane >> 4) * 2;
  const int row0g = blockIdx.y * BM;
  const int col0g = blockIdx.x * BN;

  // --- async staging addresses: 16B per lane, 128B contiguous per 8 lanes --
  // A: BM*KC floats = 1024 xb128 -> 4 per thread. B: 512 xb128 -> 2/thread.
  const float* gA[4]; unsigned sAo[4];
  const float* gB[2]; unsigned sBo[2];
  const unsigned ldsA_base = (unsigned)(uintptr_t)ldsA;   // LDS byte offset
  const unsigned ldsB_base = (unsigned)(uintptr_t)ldsB;
#pragma unroll
  for (int i = 0; i < 4; ++i) {
    int idx = tid + 256 * i, r = idx >> 3, s = idx & 7;
    gA[i]  = X + (size_t)(row0g + r) * NDIM + s * 4;
    sAo[i] = ldsA_base + (unsigned)(r * KPAD + s * 4) * 4u;
  }
#pragma unroll
  for (int i = 0; i < 2; ++i) {
    int idx = tid + 256 * i, r = idx >> 3, s = idx & 7;
    gB[i]  = X + (size_t)(col0g + r) * NDIM + s * 4;
    sBo[i] = ldsB_base + (unsigned)(r * KPAD + s * 4) * 4u;
  }

  // --- per-wave fragment offsets within an LDS buffer ---------------------
  const int fa0 = (wm * 32 + l16) * KPAD + koff;
  const int fa1 = fa0 + 16 * KPAD;
  const int fb0 = (wn * 32 + l16) * KPAD + koff;
  const int fb1 = fb0 + 16 * KPAD;

  v8f acc00 = {}, acc01 = {}, acc10 = {}, acc11 = {};

  // prologue: async-copy chunk 0 -> buffer 0
#pragma unroll
  for (int i = 0; i < 4; ++i) async_copy_b128(sAo[i], gA[i]);
#pragma unroll
  for (int i = 0; i < 2; ++i) async_copy_b128(sBo[i], gB[i]);
  wait_async0();
  __syncthreads();

  for (int c = 0; c < NCHUNK; ++c) {
    // issue next chunk's async copies into the other buffer (overlaps with
    // the entire compute phase below; waited only at the end)
    if (c + 1 < NCHUNK) {
      const int kc = (c + 1) * KC;
      const unsigned boff = (unsigned)(((c + 1) & 1) * (BM * KPAD) * 4);
      const unsigned boffB = (unsigned)(((c + 1) & 1) * (BN * KPAD) * 4);
#pragma unroll
      for (int i = 0; i < 4; ++i) async_copy_b128(sAo[i] + boff, gA[i] + kc);
#pragma unroll
      for (int i = 0; i < 2; ++i) async_copy_b128(sBo[i] + boffB, gB[i] + kc);
    }

    // compute on current buffer
    const float* Ab = ldsA + (c & 1) * (BM * KPAD);
    const float* Bb = ldsB + (c & 1) * (BN * KPAD);
#pragma unroll
    for (int kk = 0; kk < KC; kk += 4) {
      v2f a0 = *(const v2f*)(Ab + fa0 + kk);
      v2f a1 = *(const v2f*)(Ab + fa1 + kk);
      v2f b0 = *(const v2f*)(Bb + fb0 + kk);
      v2f b1 = *(const v2f*)(Bb + fb1 + kk);
      acc00 = __builtin_amdgcn_wmma_f32_16x16x4_f32(false, a0, false, b0,
                                                    (short)0, acc00, false, false);
      acc01 = __builtin_amdgcn_wmma_f32_16x16x4_f32(false, a0, false, b1,
                                                    (short)0, acc01, false, false);
      acc10 = __builtin_amdgcn_wmma_f32_16x16x4_f32(false, a1, false, b0,
                                                    (short)0, acc10, false, false);
      acc11 = __builtin_amdgcn_wmma_f32_16x16x4_f32(false, a1, false, b1,
                                                    (short)0, acc11, false, false);
    }

    // all of this wave's async copies must have landed before the barrier
    wait_async0();
    __syncthreads();
  }

  // --- fused epilogue: distance + mask mining -----------------------------
  const int mbase = (lane >> 4) * 8;
  const float INF = __builtin_inff();

#pragma unroll
  for (int mt = 0; mt < 2; ++mt) {
#pragma unroll
    for (int nt = 0; nt < 2; ++nt) {
      v8f acc = (mt == 0) ? ((nt == 0) ? acc00 : acc01)
                          : ((nt == 0) ? acc10 : acc11);
      const int j = col0g + wn * 32 + nt * 16 + l16;
      const float sqj = sq[j];
      const int taj = ta[j];
      const int tbj = tb[j];
#pragma unroll
      for (int r = 0; r < 8; ++r) {
        const int i = row0g + wm * 32 + mt * 16 + mbase + r;
        const float d2   = sq[i] + sqj - 2.0f * acc[r];
        const float dist = sqrtf(fmaxf(d2, 1e-12f));
        const bool  ma   = (ta[i] == taj);
        const bool  mb   = (tb[i] == tbj);
        const bool  s_ab = ma && mb;
        const bool  s_or = ma != mb;
        const bool  d_ab = (!ma) && (!mb);

        float cmaxab = s_ab ? dist : 0.f;
        float cminab = d_ab ? dist : INF;
        float cmaxor = s_or ? dist : 0.f;
        float cminor = s_or ? dist : INF;
#pragma unroll
        for (int off = 8; off > 0; off >>= 1) {
          cmaxab = fmaxf(cmaxab, __shfl_xor(cmaxab, off, 16));
          cminab = fminf(cminab, __shfl_xor(cminab, off, 16));
          cmaxor = fmaxf(cmaxor, __shfl_xor(cmaxor, off, 16));
          cminor = fminf(cminor, __shfl_xor(cminor, off, 16));
        }
        if (l16 == 0) {
          if (cmaxab > 0.f) atomicMax(&hp[i], __float_as_uint(cmaxab));
          if (cminab < INF) atomicMin(&hn[i], __float_as_uint(cminab));
          if (cmaxor > 0.f) {
            atomicMax(&mo[i], __float_as_uint(cmaxor));
            atomicMin(&mn[i], __float_as_uint(cminor));
            has[i] = 1u;
          }
        }
      }
    }
  }
}

// ---------------------------------------------------------------------------
// Three MarginRankingLosses -> single scalar.
// ---------------------------------------------------------------------------
__global__ __launch_bounds__(256) void finalize(
    const unsigned* __restrict__ hp, const unsigned* __restrict__ hn,
    const unsigned* __restrict__ mo, const unsigned* __restrict__ mn,
    const unsigned* __restrict__ has, const int* __restrict__ epoch,
    float* __restrict__ out) {
  __shared__ float r1[256], r2[256], r3[256];
  const bool e = (epoch[0] > 50);
  float a1 = 0.f, a2 = 0.f, a3 = 0.f;
  for (int i = threadIdx.x; i < NROWS; i += 256) {
    const float hpv = __uint_as_float(hp[i]);
    const float hnv = __uint_as_float(hn[i]);
    const bool  uo  = e && (has[i] != 0u);
    const float sp  = uo ? __uint_as_float(mo[i]) : hpv;
    const float sn  = uo ? __uint_as_float(mn[i]) : hnv;
    a1 += fmaxf(0.f, hpv - hnv + MARGIN);
    a2 += fmaxf(0.f, sp  - hnv + MARGIN);
    a3 += fmaxf(0.f, hpv - sn  + MARGIN);
  }
  r1[threadIdx.x] = a1; r2[threadIdx.x] = a2; r3[threadIdx.x] = a3;
  __syncthreads();
  for (int off = 128; off > 0; off >>= 1) {
    if (threadIdx.x < off) {
      r1[threadIdx.x] += r1[threadIdx.x + off];
      r2[threadIdx.x] += r2[threadIdx.x + off];
      r3[threadIdx.x] += r3[threadIdx.x + off];
    }
    __syncthreads();
  }
  if (threadIdx.x == 0)
    out[0] = (r1[0] + r2[0] + r3[0]) * (1.0f / (float)NROWS);
}

// ---------------------------------------------------------------------------
extern "C" void kernel_launch(void* const* d_in, const int* in_sizes, int n_in,
                              void* d_out, int out_size, void* d_ws,
                              size_t ws_size, hipStream_t stream) {
  (void)in_sizes; (void)n_in; (void)out_size; (void)ws_size;
  const float* X     = (const float*)d_in[0];  // [4096, 2048] f32
  const int*   ta    = (const int*)d_in[1];    // [4096]
  const int*   tb    = (const int*)d_in[2];    // [4096]
  /* d_in[3] = lam : unused by the reference loss */
  const int*   epoch = (const int*)d_in[4];    // scalar
  float*       out   = (float*)d_out;

  char* ws = (char*)d_ws;
  float*    sq  = (float*)ws;                  // 4096 f32
  unsigned* hp  = (unsigned*)(ws + 16384);     // 4096 u32 each below
  unsigned* hn  = hp + NROWS;
  unsigned* mo  = hn + NROWS;
  unsigned* mn  = mo + NROWS;
  unsigned* has = mn + NROWS;

  init_stats<<<NROWS / 256, 256, 0, stream>>>(hp, hn, mo, mn, has);
  sqnorm<<<NROWS, 256, 0, stream>>>(X, sq);
  dim3 grid(NROWS / BN, NROWS / BM);  // (cols/64, rows/128)
  gram_stats<<<grid, 256, 0, stream>>>(X, sq, ta, tb, hp, hn, mo, mn, has);
  finalize<<<1, 256, 0, stream>>>(hp, hn, mo, mn, has, epoch, out);
}